// SS2D_24240795418813
// MI455X (gfx1250) — compile-verified
//
#include <hip/hip_runtime.h>
#include <hip/hip_bf16.h>
#include <math.h>

#define B_   4
#define H_   56
#define W_   56
#define L_   3136          // H_*W_
#define DM   96            // d_model
#define DIN  192           // d_inner
#define NST  16            // d_state
#define RNK  6             // dt rank
#define KD   4             // directions
#define C38  38            // rank + 2*N
#define C40  40            // padded row width for xdblT
#define M_TOT (B_*L_)      // 12544 tokens
#define NCH  16            // scan chunks per chain
#define CLEN 196           // chunk length (NCH*CLEN == L_)

typedef float v2f __attribute__((ext_vector_type(2)));
typedef float v8f __attribute__((ext_vector_type(8)));

__device__ __forceinline__ float siluf(float x) {
    return x * (1.f / (1.f + __expf(-x)));
}

// map scan position l of direction k -> HW-linear spatial offset
__device__ __forceinline__ int gatherOff(int k, int l) {
    if (k == 0) return l;
    if (k == 2) return L_ - 1 - l;
    int lp = (k == 1) ? l : (L_ - 1 - l);
    int hq = lp % H_;
    int wq = lp / H_;
    return hq * W_ + wq;
}

// ---------------------------------------------------------------- zero init
__global__ __launch_bounds__(256) void k_zero(float* __restrict__ p, int n) {
    int i = blockIdx.x * 256 + threadIdx.x;
    if (i < n) p[i] = 0.f;
}

// ---------------------------------------------------------------- in_proj GEMM (WMMA f32 16x16x4)
// X[M,96] * in_proj_w[384,96]^T -> cols<192: xcpre[B,DIN,H,W]; cols>=192: silu -> zbuf[M,DIN]
__global__ __launch_bounds__(256) void k_inproj(const float* __restrict__ X,
                                                const float* __restrict__ Wq,
                                                float* __restrict__ xcpre,
                                                float* __restrict__ zbuf) {
    int wave = threadIdx.x >> 5, lane = threadIdx.x & 31;
    int tile = blockIdx.x * 8 + wave;          // 784*24 tiles
    int mtile = tile / 24, ntile = tile % 24;
    int m0 = mtile * 16, n0 = ntile * 16;
    int hlf = lane >> 4, lr = lane & 15;
    const float* arow = X  + (size_t)(m0 + lr) * DM + 2 * hlf;
    const float* brow = Wq + (size_t)(n0 + lr) * DM + 2 * hlf;
    v8f c = {};
    for (int kk = 0; kk < 24; ++kk) {
        v2f a = *(const v2f*)(arow + kk * 4);
        v2f b = *(const v2f*)(brow + kk * 4);
        c = __builtin_amdgcn_wmma_f32_16x16x4_f32(false, a, false, b, (short)0, c, false, false);
    }
    int col = n0 + lr;
    int bidx = m0 / L_;                        // tiles never cross batch (3136 % 16 == 0)
#pragma unroll
    for (int r = 0; r < 8; ++r) {
        int m = m0 + r + 8 * hlf;
        float val = c[r];
        if (col < DIN) {
            int l = m - bidx * L_;
            xcpre[(size_t)(bidx * DIN + col) * L_ + l] = val;
        } else {
            zbuf[(size_t)m * DIN + (col - DIN)] = siluf(val);
        }
    }
}

// ---------------------------------------------------------------- depthwise 3x3 conv + bias + SiLU
__global__ __launch_bounds__(256) void k_conv(const float* __restrict__ xcpre,
                                              const float* __restrict__ cw,
                                              const float* __restrict__ cb,
                                              float* __restrict__ xconv) {
    int idx = blockIdx.x * 256 + threadIdx.x;
    if (idx >= B_ * DIN * L_) return;
    int w = idx % W_;
    int h = (idx / W_) % H_;
    int d = (idx / L_) % DIN;
    int b = idx / (DIN * L_);
    const float* src = xcpre + (size_t)(b * DIN + d) * L_;
    float acc = cb[d];
#pragma unroll
    for (int i = 0; i < 3; ++i) {
        int hh = h + i - 1;
        if ((unsigned)hh >= (unsigned)H_) continue;
#pragma unroll
        for (int j = 0; j < 3; ++j) {
            int wv = w + j - 1;
            if ((unsigned)wv >= (unsigned)W_) continue;
            acc = fmaf(cw[d * 9 + i * 3 + j], src[hh * W_ + wv], acc);
        }
    }
    xconv[idx] = siluf(acc);
}

// ---------------------------------------------------------------- x_dbl GEMM (WMMA f32 16x16x4)
// per (b,k): x_proj_weight[k][38,192] * xs[b,k][192,L] -> xdblT[b,k][L,40] (transposed, padded)
__global__ __launch_bounds__(256) void k_xdbl(const float* __restrict__ xconv,
                                              const float* __restrict__ xpw,
                                              float* __restrict__ xdblT) {
    int wave = threadIdx.x >> 5, lane = threadIdx.x & 31;
    int tile = blockIdx.x * 8 + wave;          // 16*3*196 tiles
    int ltile = tile % 196;
    int ctile = (tile / 196) % 3;
    int bk    = tile / 588;
    int b = bk >> 2, k = bk & 3;
    int l0 = ltile * 16, c0 = ctile * 16;
    int hlf = lane >> 4, lr = lane & 15;
    int c = c0 + lr;
    float amask = (c < C38) ? 1.f : 0.f;
    const float* arow = xpw + (size_t)(k * C38 + (c < C38 ? c : C38 - 1)) * DIN + 2 * hlf;
    int l = l0 + lr;
    int off = gatherOff(k, l);
    const float* xb = xconv + (size_t)(b * DIN) * L_ + off;
    v8f cc = {};
    for (int kk = 0; kk < 48; ++kk) {
        v2f a = *(const v2f*)(arow + kk * 4);
        a.x *= amask; a.y *= amask;
        int d0 = kk * 4 + 2 * hlf;
        v2f bb;
        bb.x = xb[(size_t)d0 * L_];
        bb.y = xb[(size_t)(d0 + 1) * L_];
        cc = __builtin_amdgcn_wmma_f32_16x16x4_f32(false, a, false, bb, (short)0, cc, false, false);
    }
    float* orow = xdblT + ((size_t)bk * L_ + l) * C40;
#pragma unroll
    for (int r = 0; r < 8; ++r) {
        int crow = c0 + r + 8 * hlf;
        if (crow < C38) orow[crow] = cc[r];
    }
}

// ---------------------------------------------------------------- chunked selective scan
// One 16-lane group per (b,k,d) chain-chunk, lane = state index n; 2 chains per wave.
// PASS 0: run chunk from h=0, emit chunk-end state hend and sum of deltas (per chunk).
// PASS 1: run chunk from exact carry-in hin, emit y via 16-lane reduction + atomic merge.
template <int PASS>
__global__ __launch_bounds__(256) void k_scan_pass(const float* __restrict__ xconv,
                                                   const float* __restrict__ xdblT,
                                                   const float* __restrict__ dtw,
                                                   const float* __restrict__ dtb,
                                                   const float* __restrict__ Alogs,
                                                   const float* __restrict__ Dvec,
                                                   float* __restrict__ ymerge,
                                                   float* __restrict__ hend,
                                                   float* __restrict__ sumd,
                                                   const float* __restrict__ hin) {
    int wave = threadIdx.x >> 5, lane = threadIdx.x & 31;
    int wv = blockIdx.x * 8 + wave;            // 0 .. 1536*NCH-1
    int chunk = wv % NCH;
    int cpair = wv / NCH;                      // 0..1535
    int g = lane >> 4, n = lane & 15;
    int dpair = cpair % 96;
    int k = (cpair / 96) % 4;
    int b = cpair / 384;
    int d = dpair * 2 + g;
    int kd = k * DIN + d;
    int chain = (b * 4 + k) * DIN + d;         // 0..3071
    float An   = -__expf(Alogs[(size_t)kd * NST + n]);
    float bias = dtb[kd];
    float Dkd  = Dvec[kd];
    float dw[RNK];
#pragma unroll
    for (int r = 0; r < RNK; ++r) dw[r] = dtw[(size_t)kd * RNK + r];
    const float* xrow = xconv + (size_t)(b * DIN + d) * L_;
    const float* xd   = xdblT + (size_t)(b * 4 + k) * L_ * C40;
    float* yb = ymerge + (size_t)b * L_ * DIN + d;
    int l0 = chunk * CLEN;
    float h  = (PASS == 0) ? 0.f : hin[((size_t)chain * NCH + chunk) * NST + n];
    float sd = 0.f;
    // direction traversal state at l0
    int hh = 0, ww = 0, dh = 1;
    if (k & 1) {
        int lp = (k == 1) ? l0 : (L_ - 1 - l0);
        hh = lp % H_;
        ww = lp / H_;
        dh = (k == 1) ? 1 : -1;
    }
    for (int t = 0; t < CLEN; ++t) {
        int l = l0 + t;
        int off;
        if (k == 0)      off = l;
        else if (k == 2) off = L_ - 1 - l;
        else             off = hh * W_ + ww;
        float u = xrow[off];
        const float* row = xd + (size_t)l * C40;
        float dv = bias;
#pragma unroll
        for (int r = 0; r < RNK; ++r) dv = fmaf(dw[r], row[r], dv);
        float delta = (dv > 20.f) ? dv : log1pf(__expf(dv));   // softplus
        float Bn = row[6 + n];
        h = fmaf(__expf(delta * An), h, delta * Bn * u);
        if (PASS == 0) {
            sd += delta;
        } else {
            float Cn = row[22 + n];
            float p = h * Cn;
            p += __shfl_xor(p, 1, 32);
            p += __shfl_xor(p, 2, 32);
            p += __shfl_xor(p, 4, 32);
            p += __shfl_xor(p, 8, 32);
            if (n == 0) atomicAdd(yb + (size_t)off * DIN, p + Dkd * u);
        }
        if (k & 1) {
            hh += dh;
            if ((unsigned)hh >= (unsigned)H_) { hh = (dh > 0) ? 0 : H_ - 1; ww += dh; }
        }
    }
    if (PASS == 0) {
        hend[((size_t)chain * NCH + chunk) * NST + n] = h;
        if (n == 0) sumd[(size_t)chain * NCH + chunk] = sd;
    }
}

// ---------------------------------------------------------------- carry composition across chunks
// one thread per (chain, n): h_in[j+1] = exp(An * sumd[j]) * h_in[j] + hend[j]   (exact linear combine)
__global__ __launch_bounds__(256) void k_carry(const float* __restrict__ Alogs,
                                               const float* __restrict__ hend,
                                               const float* __restrict__ sumd,
                                               float* __restrict__ hin) {
    int tid = blockIdx.x * 256 + threadIdx.x;  // 0..49151
    int n = tid & 15;
    int chain = tid >> 4;                      // 0..3071
    int d = chain % DIN;
    int k = (chain / DIN) & 3;
    int kd = k * DIN + d;
    float An = -__expf(Alogs[(size_t)kd * NST + n]);
    float hc = 0.f;
    const float* he  = hend + (size_t)chain * NCH * NST + n;
    const float* sdp = sumd + (size_t)chain * NCH;
    float* hi = hin + (size_t)chain * NCH * NST + n;
#pragma unroll
    for (int j = 0; j < NCH; ++j) {
        hi[j * NST] = hc;
        hc = fmaf(__expf(An * sdp[j]), hc, he[j * NST]);
    }
}

// ---------------------------------------------------------------- LayerNorm(D=192) * z, in place
__global__ __launch_bounds__(256) void k_ln(float* __restrict__ y,
                                            const float* __restrict__ zbuf,
                                            const float* __restrict__ gamma,
                                            const float* __restrict__ beta) {
    int wave = threadIdx.x >> 5, lane = threadIdx.x & 31;
    int rowi = blockIdx.x * 8 + wave;          // 0..12543
    float* yr = y + (size_t)rowi * DIN;
    const float* zr = zbuf + (size_t)rowi * DIN;
    float v[6];
    float s1 = 0.f, s2 = 0.f;
#pragma unroll
    for (int i = 0; i < 6; ++i) { v[i] = yr[lane + 32 * i]; s1 += v[i]; s2 += v[i] * v[i]; }
#pragma unroll
    for (int m = 1; m < 32; m <<= 1) { s1 += __shfl_xor(s1, m, 32); s2 += __shfl_xor(s2, m, 32); }
    float mean = s1 * (1.f / 192.f);
    float var  = s2 * (1.f / 192.f) - mean * mean;
    float rstd = rsqrtf(var + 1e-5f);
#pragma unroll
    for (int i = 0; i < 6; ++i) {
        int idx = lane + 32 * i;
        yr[idx] = ((v[i] - mean) * rstd * gamma[idx] + beta[idx]) * zr[idx];
    }
}

// ---------------------------------------------------------------- out_proj GEMM (WMMA f32 16x16x4)
__global__ __launch_bounds__(256) void k_outproj(const float* __restrict__ act,
                                                 const float* __restrict__ Wo,
                                                 float* __restrict__ out) {
    int wave = threadIdx.x >> 5, lane = threadIdx.x & 31;
    int tile = blockIdx.x * 8 + wave;          // 784*6 tiles
    int mtile = tile / 6, ntile = tile % 6;
    int m0 = mtile * 16, n0 = ntile * 16;
    int hlf = lane >> 4, lr = lane & 15;
    const float* arow = act + (size_t)(m0 + lr) * DIN + 2 * hlf;
    const float* brow = Wo  + (size_t)(n0 + lr) * DIN + 2 * hlf;
    v8f c = {};
    for (int kk = 0; kk < 48; ++kk) {
        v2f a = *(const v2f*)(arow + kk * 4);
        v2f b = *(const v2f*)(brow + kk * 4);
        c = __builtin_amdgcn_wmma_f32_16x16x4_f32(false, a, false, b, (short)0, c, false, false);
    }
#pragma unroll
    for (int r = 0; r < 8; ++r) {
        out[(size_t)(m0 + r + 8 * hlf) * DM + n0 + lr] = c[r];
    }
}

extern "C" void kernel_launch(void* const* d_in, const int* in_sizes, int n_in,
                              void* d_out, int out_size, void* d_ws, size_t ws_size,
                              hipStream_t stream) {
    const float* x     = (const float*)d_in[0];   // [B,H,W,96]
    const float* ipw   = (const float*)d_in[1];   // [384,96]
    const float* cw    = (const float*)d_in[2];   // [192,1,3,3]
    const float* cb    = (const float*)d_in[3];   // [192]
    const float* xpw   = (const float*)d_in[4];   // [4,38,192]
    const float* dtw   = (const float*)d_in[5];   // [4,192,6]
    const float* dtb   = (const float*)d_in[6];   // [4,192]
    const float* Alog  = (const float*)d_in[7];   // [768,16]
    const float* Ds    = (const float*)d_in[8];   // [768]
    const float* gamma = (const float*)d_in[9];   // [192]
    const float* beta  = (const float*)d_in[10];  // [192]
    const float* opw   = (const float*)d_in[11];  // [96,192]
    float* out = (float*)d_out;

    float* ws = (float*)d_ws;
    float* xcpre  = ws;                                   // B*DIN*L    = 2,408,448
    float* xconv  = xcpre  + (size_t)B_ * DIN * L_;       // B*DIN*L
    float* zbuf   = xconv  + (size_t)B_ * DIN * L_;       // M*DIN
    float* xdblT  = zbuf   + (size_t)M_TOT * DIN;         // B*K*L*40   = 2,007,040
    float* ymerge = xdblT  + (size_t)B_ * KD * L_ * C40;  // B*L*DIN
    float* hend   = ymerge + (size_t)B_ * L_ * DIN;       // 3072*NCH*NST = 786,432
    float* sumd   = hend   + (size_t)3072 * NCH * NST;    // 3072*NCH     =  49,152
    float* hin    = sumd   + (size_t)3072 * NCH;          // 3072*NCH*NST = 786,432

    int nY = B_ * L_ * DIN;
    k_zero<<<(nY + 255) / 256, 256, 0, stream>>>(ymerge, nY);
    k_inproj<<<2352, 256, 0, stream>>>(x, ipw, xcpre, zbuf);               // 784*24/8
    k_conv<<<(B_ * DIN * L_ + 255) / 256, 256, 0, stream>>>(xcpre, cw, cb, xconv);
    k_xdbl<<<1176, 256, 0, stream>>>(xconv, xpw, xdblT);                   // 16*3*196/8
    k_scan_pass<0><<<3072, 256, 0, stream>>>(xconv, xdblT, dtw, dtb, Alog, Ds,
                                             ymerge, hend, sumd, hin);     // 1536*NCH/8
    k_carry<<<192, 256, 0, stream>>>(Alog, hend, sumd, hin);               // 49152/256
    k_scan_pass<1><<<3072, 256, 0, stream>>>(xconv, xdblT, dtw, dtb, Alog, Ds,
                                             ymerge, hend, sumd, hin);
    k_ln<<<1568, 256, 0, stream>>>(ymerge, zbuf, gamma, beta);             // 12544/8
    k_outproj<<<588, 256, 0, stream>>>(ymerge, opw, out);                  // 784*6/8
}